// CondConv2d_16406775070964
// MI455X (gfx1250) — compile-verified
//
#include <hip/hip_runtime.h>

typedef __attribute__((ext_vector_type(2))) float v2f;
typedef __attribute__((ext_vector_type(8))) float v8f;

#define B_    16
#define CIN   32
#define HH    256
#define WW    256
#define COUT  32
#define NEXP  8
#define KTOT  288               // CIN * 3 * 3
#define NPAR  (COUT * KTOT)     // 9216
#define LDW   264               // padded LDS input row: interior at j=4..259 (16B aligned)
#define IN_ELTS (CIN * 3 * LDW) // 25344 floats

#if __has_builtin(__builtin_amdgcn_global_load_async_to_lds_b128)
#define HAVE_ASYNC 1
#else
#define HAVE_ASYNC 0
#endif

// Async-copy builtin parameter types (from hipcc diagnostic: it expects
// pointer-to-16B-int-vector, global AS on the source, LDS AS on the dest).
typedef __attribute__((__vector_size__(4 * sizeof(int)))) int v4i;
typedef __attribute__((address_space(1))) v4i* gv4i_p;
typedef __attribute__((address_space(3))) v4i* lv4i_p;

__device__ __forceinline__ void wait_async0() {
#if __has_builtin(__builtin_amdgcn_s_wait_asynccnt)
    __builtin_amdgcn_s_wait_asynccnt(0);
#else
    asm volatile("s_wait_asynccnt 0x0" ::: "memory");
#endif
}

// ---------------------------------------------------------------------------
// Kernel 1: W_b = routing_b @ expert_weight  (B x 9216), bias_b = routing_b @ expert_bias
// ---------------------------------------------------------------------------
__global__ void __launch_bounds__(256)
gen_weights(const float* __restrict__ routing,   // [B][NEXP]
            const float* __restrict__ ew,        // [NEXP][NPAR]
            const float* __restrict__ eb,        // [NEXP][COUT]
            float* __restrict__ wOut,            // [B][NPAR]
            float* __restrict__ bOut)            // [B][COUT]
{
    const int b   = blockIdx.x;
    const int tid = threadIdx.x;

    float r[NEXP];
#pragma unroll
    for (int e = 0; e < NEXP; ++e) r[e] = routing[b * NEXP + e];

    for (int p = tid; p < NPAR; p += 256) {
        float acc = 0.f;
#pragma unroll
        for (int e = 0; e < NEXP; ++e) acc += r[e] * ew[e * NPAR + p];
        wOut[b * NPAR + p] = acc;
    }
    if (tid < COUT) {
        float acc = 0.f;
#pragma unroll
        for (int e = 0; e < NEXP; ++e) acc += r[e] * eb[e * COUT + tid];
        bOut[b * COUT + tid] = acc;
    }
}

// ---------------------------------------------------------------------------
// Kernel 2: implicit-GEMM conv via V_WMMA_F32_16X16X4_F32.
// One block = (sample b, output row y). 8 waves, each owns a 32(M)x32(N) tile.
// Staging uses GLOBAL_LOAD_ASYNC_TO_LDS_B128 (ASYNCcnt) when available.
// ---------------------------------------------------------------------------
__global__ void __launch_bounds__(256)
condconv_wmma(const float* __restrict__ x,     // [B][CIN][H][W]
              const float* __restrict__ wAll,  // [B][COUT][KTOT]
              const float* __restrict__ bAll,  // [B][COUT]
              float* __restrict__ out)         // [B][COUT][H][W]
{
    extern __shared__ float smem[];
    float* inlds = smem;                       // IN_ELTS  : x halo rows (zero-padded)
    float* wlds  = smem + IN_ELTS;             // NPAR     : W_b row-major [COUT][KTOT]
    float* blds  = wlds + NPAR;                // COUT     : bias
    int*   offs  = (int*)(blds + COUT);        // KTOT     : k -> LDS offset table

    const int bid = blockIdx.x;
    const int b   = bid >> 8;
    const int y   = bid & 255;
    const int tid = threadIdx.x;

    const float* xb   = x + (size_t)b * CIN * HH * WW;
    const float* wsrc = wAll + (size_t)b * NPAR;

    // ---- im2col offset table: k = c*9 + kh*3 + kw -> (c*3+kh)*LDW + kw + 3 ----
    for (int k = tid; k < KTOT; k += 256) {
        int c  = k / 9;
        int rem = k - c * 9;
        int kh = rem / 3;
        int kw = rem - kh * 3;
        offs[k] = (c * 3 + kh) * LDW + kw + 3;
    }
    if (tid < COUT) blds[tid] = bAll[b * COUT + tid];

#if HAVE_ASYNC
    // ---- zero halo columns (j=3 and j=260) of all 96 slabs ----
    if (tid < CIN * 3 * 2) {
        int slab = tid >> 1;
        int j    = (tid & 1) ? (LDW - 4) : 3;
        inlds[slab * LDW + j] = 0.f;
    }
    // ---- zero out-of-range row slabs (only at image top/bottom) ----
    if (y == 0 || y == HH - 1) {
        int rb = (y == 0) ? 0 : 2;
        for (int i = tid; i < CIN * LDW; i += 256) {
            int c = i / LDW;
            int j = i - c * LDW;
            inlds[(c * 3 + rb) * LDW + j] = 0.f;
        }
    }
    // ---- async-stage weights: 2304 x b128, 16B aligned both sides ----
    for (int p4 = tid; p4 < NPAR / 4; p4 += 256) {
        __builtin_amdgcn_global_load_async_to_lds_b128(
            (gv4i_p)(wsrc + p4 * 4), (lv4i_p)(wlds + p4 * 4), 0, 0);
    }
    // ---- async-stage input interior: 96 slabs x 64 x b128 ----
    {
        const int y0 = y - 1;
        for (int i = tid; i < CIN * 3 * (WW / 4); i += 256) {  // 6144
            int q    = i & 63;      // 16B chunk within row
            int slab = i >> 6;      // c*3 + r
            int c    = slab / 3;
            int r    = slab - c * 3;
            int gr   = y0 + r;
            if ((unsigned)gr < HH) {
                const float* src = xb + ((size_t)c * HH + gr) * WW + q * 4;
                float*       dst = inlds + slab * LDW + 4 + q * 4;
                __builtin_amdgcn_global_load_async_to_lds_b128(
                    (gv4i_p)src, (lv4i_p)dst, 0, 0);
            }
        }
    }
    wait_async0();
#else
    // ---- fallback: VGPR-staged copy ----
    for (int p = tid; p < NPAR; p += 256) wlds[p] = wsrc[p];
    for (int idx = tid; idx < IN_ELTS; idx += 256) {
        int j  = idx % LDW;
        int cr = idx / LDW;        // c*3 + r
        int r  = cr % 3;
        int c  = cr / 3;
        int gr = y - 1 + r;
        int gc = j - 4;            // interior at j=4..259
        float v = 0.f;
        if ((unsigned)gr < HH && (unsigned)gc < WW)
            v = xb[((size_t)c * HH + gr) * WW + gc];
        inlds[idx] = v;
    }
#endif
    __syncthreads();

    // ---- per-wave GEMM tile ----
    const int lane = tid & 31;
    const int wv   = tid >> 5;       // 0..7 -> N base = wv*32
    const int lx   = lane & 15;
    const int hi   = lane >> 4;      // half-wave select (K 0,1 vs 2,3 / M 0..7 vs 8..15)
    const int xcol0 = wv * 32 + lx;  // N tile 0
    const int xcol1 = xcol0 + 16;    // N tile 1
    const int row0  = lx;            // A rows, M-tile 0
    const int row1  = 16 + lx;       // A rows, M-tile 1

    v8f c00 = {}, c01 = {}, c10 = {}, c11 = {};

#pragma unroll 4
    for (int kc = 0; kc < KTOT / 4; ++kc) {
        const int k0 = kc * 4 + hi * 2;  // even -> 8B-aligned float2 loads

        // A fragments: W_b[row][k0], W_b[row][k0+1] contiguous in row-major
        v2f a0 = *(const v2f*)(wlds + row0 * KTOT + k0);
        v2f a1 = *(const v2f*)(wlds + row1 * KTOT + k0);

        // B fragments: im2col rows k0, k0+1 at this wave's spatial columns
        const int o0 = offs[k0];
        const int o1 = offs[k0 + 1];
        v2f b0, b1;
        b0.x = inlds[o0 + xcol0];
        b0.y = inlds[o1 + xcol0];
        b1.x = inlds[o0 + xcol1];
        b1.y = inlds[o1 + xcol1];

        c00 = __builtin_amdgcn_wmma_f32_16x16x4_f32(false, a0, false, b0, (short)0, c00, false, false);
        c01 = __builtin_amdgcn_wmma_f32_16x16x4_f32(false, a0, false, b1, (short)0, c01, false, false);
        c10 = __builtin_amdgcn_wmma_f32_16x16x4_f32(false, a1, false, b0, (short)0, c10, false, false);
        c11 = __builtin_amdgcn_wmma_f32_16x16x4_f32(false, a1, false, b1, (short)0, c11, false, false);
    }

    // ---- bias + store. C/D layout: VGPR vi -> M = vi + hi*8, N = lane&15 ----
    float* ob = out + ((size_t)b * COUT * HH + y) * WW;
#pragma unroll
    for (int vi = 0; vi < 8; ++vi) {
        const int co0 = vi + hi * 8;        // M-tile 0 channel
        const int co1 = 16 + co0;           // M-tile 1 channel
        const float bias0 = blds[co0];
        const float bias1 = blds[co1];
        ob[(size_t)co0 * HH * WW + xcol0] = c00[vi] + bias0;
        ob[(size_t)co0 * HH * WW + xcol1] = c01[vi] + bias0;
        ob[(size_t)co1 * HH * WW + xcol0] = c10[vi] + bias1;
        ob[(size_t)co1 * HH * WW + xcol1] = c11[vi] + bias1;
    }
}

// ---------------------------------------------------------------------------
extern "C" void kernel_launch(void* const* d_in, const int* in_sizes, int n_in,
                              void* d_out, int out_size, void* d_ws, size_t ws_size,
                              hipStream_t stream) {
    const float* x       = (const float*)d_in[0];   // [16][32][256][256]
    const float* routing = (const float*)d_in[1];   // [16][8]
    const float* ew      = (const float*)d_in[2];   // [8][9216]
    const float* eb      = (const float*)d_in[3];   // [8][32]
    float* out = (float*)d_out;                     // [16][32][256][256]

    float* wWs = (float*)d_ws;                      // [16][9216]
    float* bWs = wWs + (size_t)B_ * NPAR;           // [16][32]

    gen_weights<<<B_, 256, 0, stream>>>(routing, ew, eb, wWs, bWs);

    const size_t shmem = (size_t)(IN_ELTS + NPAR + COUT) * sizeof(float)
                       + (size_t)KTOT * sizeof(int);   // ~140 KB -> 2 blocks/WGP
    condconv_wmma<<<B_ * HH, 256, shmem, stream>>>(x, wWs, bWs, out);
}